// iSTFT_88562225643700
// MI455X (gfx1250) — compile-verified
//
#include <hip/hip_runtime.h>

typedef __attribute__((ext_vector_type(2))) float v2f;
typedef __attribute__((ext_vector_type(8))) float v8f;

#define N_FFT    2048
#define T_FRAMES 512
#define NBATCH   4
#define PAD      1024
#define OUT_PER  261632   // N_FFT + 512*(T-1) - 2*PAD

#define BN  128           // n rows per block
#define BT  64            // t cols per block
#define BK  32            // f (K) per block
#define LDA 36            // padded LDS stride (floats); 144 B (16B multiple)
#define LDB 36
#define NIT (N_FFT / BK)  // 64 K-block iterations

// async 16B copy: global -> LDS (GLOBAL_LOAD_ASYNC_TO_LDS_B128, ASYNCcnt)
// builtin prototype (from hipcc diagnostic): (int4 AS1*, int4 AS3*, int, int)
__device__ __forceinline__ void async_cp16(void* l, const void* g) {
    typedef __attribute__((__vector_size__(16))) int vi4;
    typedef __attribute__((address_space(1))) vi4* gp_t;
    typedef __attribute__((address_space(3))) vi4* lp_t;
    __builtin_amdgcn_global_load_async_to_lds_b128(
        (gp_t)(size_t)g, (lp_t)l, 0, 0);
}

// ---------------------------------------------------------------------------
// Stage one K-block into the selected LDS buffer.
//  A tiles (cos/sin, 128x32 each): pure byte copies -> async DMA to LDS.
//  B tile  (32f x 64t complex):    de-interleave + negate Xi -> manual path.
// ---------------------------------------------------------------------------
__device__ __forceinline__ void fill_buffers(
    float* __restrict__ asc, float* __restrict__ ass,
    float* __restrict__ bsr, float* __restrict__ bsi,
    const float* __restrict__ Kcos, const float* __restrict__ Ksin,
    const float* __restrict__ X,
    int nBase, int tBase, int b, int f0, int tid)
{
    // ---- A tiles via async DMA: 128 rows x 8 chunks of 16B per matrix ----
#pragma unroll
    for (int i = 0; i < 4; ++i) {
        const int id  = tid + 256 * i;       // 0..1023
        const int row = id >> 3;             // 0..127
        const int c4  = (id & 7) << 2;       // 0,4,...,28 (floats)
        const float* gc = Kcos + (size_t)(nBase + row) * N_FFT + f0 + c4;
        const float* gs = Ksin + (size_t)(nBase + row) * N_FFT + f0 + c4;
        async_cp16(&asc[row * LDA + c4], gc);
        async_cp16(&ass[row * LDA + c4], gs);
    }
    // ---- B tile: X[b][f0+f][t][0:2] -> Bs_r[t][f]=Xr, Bs_i[t][f]=-Xi ----
    const int bF = tid >> 3;                 // 0..31
    const int bT = (tid & 7) << 3;           // 0,8,...,56
    const float2* gx = (const float2*)X
        + ((size_t)(b * N_FFT + f0 + bF) * T_FRAMES + tBase + bT);
#pragma unroll
    for (int j = 0; j < 8; ++j) {
        float2 v = gx[j];
        bsr[(bT + j) * LDB + bF] = v.x;
        bsi[(bT + j) * LDB + bF] = -v.y;
    }
}

// ---------------------------------------------------------------------------
// GEMM: frames[b][n][t] = (sum_f Xr*cos[n,f] + (-Xi)*sin[n,f]) * win[n]/N_FFT
// ---------------------------------------------------------------------------
__global__ __launch_bounds__(256)
void istft_gemm_kernel(const float* __restrict__ X,      // (B, N_FFT, T, 2)
                       const float* __restrict__ Kcos,   // (N_FFT, N_FFT) [n][f]
                       const float* __restrict__ Ksin,   // (N_FFT, N_FFT) [n][f]
                       const float* __restrict__ win,    // (N_FFT)
                       float* __restrict__ frames)       // (B, N_FFT, T)
{
    __shared__ float As_c[2 * BN * LDA];
    __shared__ float As_s[2 * BN * LDA];
    __shared__ float Bs_r[2 * BT * LDB];   // transposed: [t][f]
    __shared__ float Bs_i[2 * BT * LDB];   // holds -Xi

    const int tid   = threadIdx.x;
    const int tBase = blockIdx.x * BT;
    const int nBase = blockIdx.y * BN;
    const int b     = blockIdx.z;

    const int wave = tid >> 5;   // 0..7
    const int lane = tid & 31;
    const int wn   = wave & 3;   // wave n-offset: wn*32
    const int wt   = wave >> 2;  // wave t-offset: wt*32
    const int l15  = lane & 15;
    const int hi   = lane >> 4;  // 0: K pair {0,1}, 1: K pair {2,3}
    const int klo  = hi << 1;

    v8f acc[2][2];
#pragma unroll
    for (int i = 0; i < 2; ++i)
#pragma unroll
        for (int j = 0; j < 2; ++j)
            acc[i][j] = (v8f){0.f, 0.f, 0.f, 0.f, 0.f, 0.f, 0.f, 0.f};

    const int ar0 = (wn * 32 + l15) * LDA;
    const int ar1 = ar0 + 16 * LDA;
    const int br0 = (wt * 32 + l15) * LDB;
    const int br1 = br0 + 16 * LDB;

    // prologue: fill buffer 0 for f0 = 0
    fill_buffers(As_c, As_s, Bs_r, Bs_i, Kcos, Ksin, X, nBase, tBase, b, 0, tid);

    for (int it = 0; it < NIT; ++it) {
        const int cur = it & 1;
        float* asc = As_c + cur * (BN * LDA);
        float* ass = As_s + cur * (BN * LDA);
        float* bsr = Bs_r + cur * (BT * LDB);
        float* bsi = Bs_i + cur * (BT * LDB);

        // drain this wave's async fills, then make all fills visible
        asm volatile("s_wait_asynccnt 0x0" ::: "memory");
        __syncthreads();

        // prefetch next K-block into the other buffer (overlaps the WMMAs)
        if (it + 1 < NIT) {
            const int nxt = cur ^ 1;
            fill_buffers(As_c + nxt * (BN * LDA), As_s + nxt * (BN * LDA),
                         Bs_r + nxt * (BT * LDB), Bs_i + nxt * (BT * LDB),
                         Kcos, Ksin, X, nBase, tBase, b, (it + 1) * BK, tid);
        }

        // ---- 32x32 per-wave accumulate via v_wmma_f32_16x16x4_f32 -----
#pragma unroll
        for (int kk = 0; kk < BK; kk += 4) {
            const int ko = kk + klo;
            // cos path with Xr
            v2f a0 = *(const v2f*)&asc[ar0 + ko];
            v2f a1 = *(const v2f*)&asc[ar1 + ko];
            v2f b0 = *(const v2f*)&bsr[br0 + ko];
            v2f b1 = *(const v2f*)&bsr[br1 + ko];
            acc[0][0] = __builtin_amdgcn_wmma_f32_16x16x4_f32(false, a0, false, b0, (short)0, acc[0][0], false, false);
            acc[0][1] = __builtin_amdgcn_wmma_f32_16x16x4_f32(false, a0, false, b1, (short)0, acc[0][1], false, false);
            acc[1][0] = __builtin_amdgcn_wmma_f32_16x16x4_f32(false, a1, false, b0, (short)0, acc[1][0], false, false);
            acc[1][1] = __builtin_amdgcn_wmma_f32_16x16x4_f32(false, a1, false, b1, (short)0, acc[1][1], false, false);
            // sin path with -Xi
            v2f c0 = *(const v2f*)&ass[ar0 + ko];
            v2f c1 = *(const v2f*)&ass[ar1 + ko];
            v2f d0 = *(const v2f*)&bsi[br0 + ko];
            v2f d1 = *(const v2f*)&bsi[br1 + ko];
            acc[0][0] = __builtin_amdgcn_wmma_f32_16x16x4_f32(false, c0, false, d0, (short)0, acc[0][0], false, false);
            acc[0][1] = __builtin_amdgcn_wmma_f32_16x16x4_f32(false, c0, false, d1, (short)0, acc[0][1], false, false);
            acc[1][0] = __builtin_amdgcn_wmma_f32_16x16x4_f32(false, c1, false, d0, (short)0, acc[1][0], false, false);
            acc[1][1] = __builtin_amdgcn_wmma_f32_16x16x4_f32(false, c1, false, d1, (short)0, acc[1][1], false, false);
        }
        __syncthreads();   // all reads of `cur` done before it is refilled
    }

    // ---- epilogue: scale by win[n]/N_FFT and store frames -------------
    const float invN = 1.0f / (float)N_FFT;
#pragma unroll
    for (int i = 0; i < 2; ++i) {
#pragma unroll
        for (int j = 0; j < 2; ++j) {
            const int tcol = tBase + wt * 32 + j * 16 + l15;
#pragma unroll
            for (int r = 0; r < 8; ++r) {
                const int n = nBase + wn * 32 + i * 16 + r + hi * 8;
                const float val = acc[i][j][r] * win[n] * invN;
                frames[((size_t)(b * N_FFT) + n) * T_FRAMES + tcol] = val;
            }
        }
    }
}

// ---------------------------------------------------------------------------
// Overlap-add + window-sum-square normalize + center trim.
// ---------------------------------------------------------------------------
__global__ __launch_bounds__(256)
void istft_ola_kernel(const float* __restrict__ frames,
                      const float* __restrict__ win,
                      float* __restrict__ out)
{
    const int s1 = blockIdx.x * blockDim.x + threadIdx.x;
    if (s1 >= OUT_PER) return;
    const int b = blockIdx.y;
    const int s = s1 + PAD;

    int tmax = s >> 9;               // floor(s/512)
    if (tmax > T_FRAMES - 1) tmax = T_FRAMES - 1;
    const int tmin = (s >= N_FFT) ? (((s - N_FFT) >> 9) + 1) : 0;

    float y  = 0.0f;
    float ws = 0.0f;
    for (int t = tmin; t <= tmax; ++t) {
        const int n = s - (t << 9);                       // 0..2047
        y += frames[((size_t)(b * N_FFT) + n) * T_FRAMES + t];
        const float w = win[n];
        ws += w * w;
    }
    if (ws > 1e-10f) y /= ws;
    out[(size_t)b * OUT_PER + s1] = y;
}

extern "C" void kernel_launch(void* const* d_in, const int* in_sizes, int n_in,
                              void* d_out, int out_size, void* d_ws, size_t ws_size,
                              hipStream_t stream) {
    (void)in_sizes; (void)n_in; (void)out_size; (void)ws_size;
    const float* X  = (const float*)d_in[0];   // (4, 2048, 512, 2)
    const float* Kc = (const float*)d_in[1];   // (2048, 2048)
    const float* Ks = (const float*)d_in[2];   // (2048, 2048)
    const float* W  = (const float*)d_in[3];   // (2048,)
    float* out    = (float*)d_out;             // (4, 261632)
    float* frames = (float*)d_ws;              // (4, 2048, 512) = 16.8 MB scratch

    dim3 grid1(T_FRAMES / BT, N_FFT / BN, NBATCH);   // (8, 16, 4)
    istft_gemm_kernel<<<grid1, dim3(256), 0, stream>>>(X, Kc, Ks, W, frames);

    dim3 grid2((OUT_PER + 255) / 256, NBATCH);
    istft_ola_kernel<<<grid2, dim3(256), 0, stream>>>(frames, W, out);
}